// ConvTranspose1d_721554505812
// MI455X (gfx1250) — compile-verified
//
#include <hip/hip_runtime.h>

typedef __attribute__((ext_vector_type(2))) float v2f;
typedef __attribute__((ext_vector_type(4))) float v4f;
typedef __attribute__((ext_vector_type(8))) float v8f;

#define CIN   64
#define COUT  64
#define KW    8
#define LIN   32768
#define LOUT  (LIN + KW - 1)          // 32775
#define KK    (CIN * KW)              // 512 reduction length
#define TB    128                     // t-columns per block (8 waves x 16)
#define XROW  136                     // padded halo row: 135 valid + 1 pad
#define CPH   32                      // channels staged per phase

__global__ __launch_bounds__(256)
void convt1d_wmma_f32(const float* __restrict__ x,
                      const float* __restrict__ w,
                      const float* __restrict__ bias,
                      float* __restrict__ out)
{
    // LDS:
    //  sW4[(c*2+hf)*16+ln] = (w[m0+ln, c, 4hf], .., w[m0+ln, c, 4hf+3])  -> 32 KB
    //  sX: time-REVERSED x rows, 32 channels + halo per phase            -> 17 KB
    __shared__ v4f   sW4[CIN * 2 * 16];
    __shared__ float sX[CPH * XROW];
    __shared__ float sB[COUT];

    const int tid = threadIdx.x;
    const int bt0 = blockIdx.x * TB;   // base t of this supertile
    const int m0  = blockIdx.y * 16;   // base cout of this tile
    const int b   = blockIdx.z;

    // ---- stage weight tile: 2048 float4, straight 16B copies
    {
        const v4f* wv4 = (const v4f*)w;          // w[o][c][j], 2 v4f per (o,c)
        for (int e = tid; e < CIN * 2 * 16; e += 256) {
            int p  = e >> 5;                     // channel c
            int h2 = (e >> 4) & 1;               // hf
            int l2 = e & 15;                     // ln -> m = m0 + ln
            sW4[e] = wv4[(size_t)(m0 + l2) * (KK / 4) + p * 2 + h2];
        }
    }
    if (tid < COUT) sB[tid] = bias[tid];

    const int lane = tid & 31;
    const int wv32 = tid >> 5;         // wave id 0..7
    const int hf   = lane >> 4;        // half: 0 or 1
    const int ln   = lane & 15;        // N column within tile
    const int tloc = wv32 * 16 + ln;   // local t in [0,127]

    v8f acc = {};

    for (int phase = 0; phase < 2; ++phase) {
        __syncthreads();               // protect sX overwrite / first fills
        // ---- stage x channels [phase*32 .. +32) TIME-REVERSED with halo:
        //      sX[cl][i] = x[b, c, bt0 + 127 - i], zero outside [0, LIN)
        for (int e = tid; e < CPH * XROW; e += 256) {
            int cl = e / XROW;
            int i  = e - cl * XROW;
            int c  = phase * CPH + cl;
            int gt = bt0 + (TB - 1) - i;          // reversed time index
            float v = 0.0f;
            if (i <= TB + KW - 2 && gt >= 0 && gt < LIN)
                v = x[(size_t)b * CIN * LIN + (size_t)c * LIN + gt];
            sX[e] = v;
        }
        __syncthreads();

        // ---- 64 WMMA steps this phase, 32 channels x 2 steps:
        //      even step rows j = (4hf, 4hf+1); odd step rows j = (4hf+2, 4hf+3)
        const v4f*   aBase = sW4 + (size_t)(phase * CPH) * 32 + hf * 16 + ln;
        const float* bBase = sX + (TB - 1) - tloc + 4 * hf;   // -> x[t-4hf]

        #pragma unroll 8
        for (int sp = 0; sp < CPH; ++sp) {
            v4f av = aBase[(size_t)sp * 32];          // one b128: both A frags
            const float* bp = bBase + (size_t)sp * XROW;

            v2f a0 = { av.x, av.y };
            v2f b0 = { bp[0], bp[1] };                // (x[t-j0], x[t-j0-1])
            acc = __builtin_amdgcn_wmma_f32_16x16x4_f32(
                      false, a0, false, b0, (short)0, acc, false, false);

            v2f a1 = { av.z, av.w };
            v2f b1 = { bp[2], bp[3] };                // (x[t-j0-2], x[t-j0-3])
            acc = __builtin_amdgcn_wmma_f32_16x16x4_f32(
                      false, a1, false, b1, (short)0, acc, false, false);
        }
    }

    // ---- epilogue: bias + store per 16x16 f32 C/D layout
    int t = bt0 + tloc;
    if (t < LOUT) {
        #pragma unroll
        for (int r = 0; r < 8; ++r) {
            int o = m0 + r + 8 * hf;
            out[(size_t)b * COUT * LOUT + (size_t)o * LOUT + t] = acc[r] + sB[o];
        }
    }
}

extern "C" void kernel_launch(void* const* d_in, const int* in_sizes, int n_in,
                              void* d_out, int out_size, void* d_ws, size_t ws_size,
                              hipStream_t stream) {
    const float* x    = (const float*)d_in[0];
    const float* wgt  = (const float*)d_in[1];
    const float* bias = (const float*)d_in[2];
    float* out = (float*)d_out;

    const int nt = (LOUT + TB - 1) / TB;   // 257 t-supertiles
    dim3 grid(nt, COUT / 16, 16);          // (257, 4, 16)
    dim3 block(256);
    convt1d_wmma_f32<<<grid, block, 0, stream>>>(x, wgt, bias, out);
}